// InterfaceBoundaryLoss_23347442221431
// MI455X (gfx1250) — compile-verified
//
#include <hip/hip_runtime.h>

// ---------------------------------------------------------------------------
// InterfaceBoundaryLoss for MI455X (gfx1250, wave32)
//   Pass 1: branch-free upwind-FD gather (cndmask offset + sign flip,
//           8 loads/sample, FD scale as multiply-by-191 not IEEE divide),
//           per-wave 32->2 collapse via V_WMMA_F32_16X16X4_F32 ones-matrix
//           row-sum, 2D grid (y=batch, x covers boundary points coalesced)
//   Pass 2: single wave32 reduces block partials with the same WMMA trick
// ---------------------------------------------------------------------------

#define GRID_N   192
#define NBATCH   4
#define E_IN_C   1.0f
#define E_OUT_C  80.0f
#define INV_DX   ((float)(GRID_N - 1))   // 1/DX = 191 exactly (DX = 1/(N-1))

typedef float v2f __attribute__((ext_vector_type(2)));
typedef float v8f __attribute__((ext_vector_type(8)));

// Collapse 32 lane partials with one WMMA: A[m][0]=acc[m], A[m][2]=acc[m+16],
// B = ones(4x16) -> D[m][n] = rowsum(A[m]). Lanes 0..15 hold rows 0..7 in the
// 8 accumulator VGPRs, lanes 16..31 hold rows 8..15. Returned t: lanes 0..15
// carry sum(acc[0..7],acc[16..23]), lanes 16..31 carry the other half.
// Requires full EXEC (call only from convergent code).
__device__ __forceinline__ float wave_halfsum_wmma(float acc) {
    v2f a;  a[0] = acc;  a[1] = 0.0f;
    v2f bm; bm[0] = 1.0f; bm[1] = 1.0f;
    v8f c = {};
    v8f d = __builtin_amdgcn_wmma_f32_16x16x4_f32(
        false, a, false, bm, (short)0, c, false, false);
    return d[0] + d[1] + d[2] + d[3] + d[4] + d[5] + d[6] + d[7];
}

__global__ void boundary_loss_partial(
    const float* __restrict__ fin, const float* __restrict__ fout,
    const int*   __restrict__ xi,  const int*   __restrict__ yi,
    const int*   __restrict__ zi,
    const float* __restrict__ nxv, const float* __restrict__ nyv,
    const float* __restrict__ nzv,
    float* __restrict__ partials, int P)
{
    const int SX = GRID_N * GRID_N, SY = GRID_N, SZ = 1;

    const int p = blockIdx.x * blockDim.x + threadIdx.x;
    const int b = blockIdx.y;                      // batch from grid dim

    float acc = 0.0f;
    if (p < P) {
        int x = xi[p], y = yi[p], z = zi[p];
        float nx = nxv[p], ny = nyv[p], nz = nzv[p];

        const size_t vol  = (size_t)GRID_N * GRID_N * GRID_N;
        const size_t base = ((size_t)x * GRID_N + (size_t)y) * GRID_N + (size_t)z;
        const float* pin  = fin  + (size_t)b * vol + base;
        const float* pout = fout + (size_t)b * vol + base;

        float vi = pin[0];
        float vo = pout[0];

        // Branch-free upwind: in-neighbor at s, out-neighbor at -s,
        //   g_in = sgn*(vi - nbr_in)*INV_DX, g_out = sgn*(nbr_out - vo)*INV_DX
        int   sx  = (nx > 0.0f) ? -SX : SX;
        float sgx = (nx > 0.0f) ? 1.0f : -1.0f;
        int   sy  = (ny > 0.0f) ? -SY : SY;
        float sgy = (ny > 0.0f) ? 1.0f : -1.0f;
        int   sz  = (nz > 0.0f) ? -SZ : SZ;
        float sgz = (nz > 0.0f) ? 1.0f : -1.0f;

        float gxi = sgx * (vi - pin[sx])   * INV_DX;
        float gxo = sgx * (pout[-sx] - vo) * INV_DX;
        float gyi = sgy * (vi - pin[sy])   * INV_DX;
        float gyo = sgy * (pout[-sy] - vo) * INV_DX;
        float gzi = sgz * (vi - pin[sz])   * INV_DX;
        float gzo = sgz * (pout[-sz] - vo) * INV_DX;

        float ndi = gxi * nx + gyi * ny + gzi * nz;
        float ndo = gxo * nx + gyo * ny + gzo * nz;

        float d0 = vi - vo;
        float d1 = E_IN_C * ndi - E_OUT_C * ndo;
        acc = d0 * d0 + d1 * d1;               // both means share 1/M -> fuse
    }

    // Convergent here: every wave has full EXEC -> WMMA wave reduction.
    float t = wave_halfsum_wmma(acc);

    __shared__ float wsum[16];                 // 2 half-sums per wave, 8 waves
    const int wave = threadIdx.x >> 5;
    const int lane = threadIdx.x & 31;
    if ((lane & 15) == 0)                      // lanes 0 and 16 carry halves
        wsum[(wave << 1) | (lane >> 4)] = t;
    __syncthreads();

    if (threadIdx.x == 0) {
        float s = 0.0f;
        #pragma unroll
        for (int i = 0; i < 16; ++i) s += wsum[i];
        partials[blockIdx.y * gridDim.x + blockIdx.x] = s;
    }
}

__global__ void boundary_loss_final(const float* __restrict__ partials,
                                    int nparts, float* __restrict__ out,
                                    float invM)
{
    const int lane = threadIdx.x;              // exactly one wave32
    float acc = 0.0f;
    for (int i = lane; i < nparts; i += 32) acc += partials[i];

    float t = wave_halfsum_wmma(acc);

    __shared__ float red[32];
    red[lane] = t;
    __syncthreads();
    if (lane == 0) out[0] = (red[0] + red[16]) * invM;   // WEIGHT == 1
}

extern "C" void kernel_launch(void* const* d_in, const int* in_sizes, int n_in,
                              void* d_out, int out_size, void* d_ws, size_t ws_size,
                              hipStream_t stream)
{
    const float* fin  = (const float*)d_in[0];
    const float* fout = (const float*)d_in[1];
    const int*   xi   = (const int*)d_in[2];
    const int*   yi   = (const int*)d_in[3];
    const int*   zi   = (const int*)d_in[4];
    const float* nx   = (const float*)d_in[5];
    const float* ny   = (const float*)d_in[6];
    const float* nz   = (const float*)d_in[7];

    const int P = in_sizes[2];
    const int threads  = 256;
    const int nblocksP = (P + threads - 1) / threads;
    const int nparts   = NBATCH * nblocksP;

    float* partials = (float*)d_ws;            // nparts floats << ws_size

    dim3 grid(nblocksP, NBATCH);
    boundary_loss_partial<<<grid, threads, 0, stream>>>(
        fin, fout, xi, yi, zi, nx, ny, nz, partials, P);

    boundary_loss_final<<<1, 32, 0, stream>>>(
        partials, nparts, (float*)d_out, 1.0f / (float)(NBATCH * P));
}